// RDB_2001454760241
// MI455X (gfx1250) — compile-verified
//
#include <hip/hip_runtime.h>
#include <hip/hip_bf16.h>

typedef float v2f __attribute__((ext_vector_type(2)));
typedef float v8f __attribute__((ext_vector_type(8)));

#define NN     20000
#define NEDGE  320000
#define ETOT   (NEDGE + NN)            /* 340000, multiple of 32 */
#define INV_PI 0.3183098861837907f

__device__ __forceinline__ float selu_f(float v) {
    const float alpha = 1.6732632423543772f;
    const float scale = 1.0507009873554805f;
    return v > 0.0f ? scale * v : scale * alpha * (__expf(v) - 1.0f);
}

// ---------------------------------------------------------------- edge feats
__global__ __launch_bounds__(256)
void feats_kernel(const float* __restrict__ pos,
                  const int* __restrict__ srcI,
                  const int* __restrict__ dstI,
                  float* __restrict__ feats) {
    int e = blockIdx.x * blockDim.x + threadIdx.x;
    if (e >= ETOT) return;
    float rx = 0.f, ry = 0.f, rz = 0.f;
    if (e < NEDGE) {
        int s = srcI[e], d = dstI[e];
        rx = pos[d * 3 + 0] - pos[s * 3 + 0];
        ry = pos[d * 3 + 1] - pos[s * 3 + 1];
        rz = pos[d * 3 + 2] - pos[s * 3 + 2];
    }
    float sq = rx * rx + ry * ry + rz * rz;
    float rho = 0.f, theta = 0.f, phi = 0.f;
    if (sq != 0.0f) {
        rho = sqrtf(sq);
        theta = atan2f(ry, rx) * INV_PI;
        float zr = fminf(fmaxf(rz / rho, -1.0f), 1.0f);
        phi = asinf(zr) * INV_PI;
    }
    feats[e * 3 + 0] = rho;
    feats[e * 3 + 1] = theta;
    feats[e * 3 + 2] = phi;
}

// ---------------------------------------------------------------- node-side
__global__ __launch_bounds__(256)
void init_agg_kernel(float* __restrict__ agg, const float* __restrict__ bias) {
    int i = blockIdx.x * blockDim.x + threadIdx.x;
    if (i < NN * 16) agg[i] = bias[i & 15];
}

__global__ __launch_bounds__(256)
void act_kernel(const float* __restrict__ agg, float* __restrict__ out) {
    int i = blockIdx.x * blockDim.x + threadIdx.x;
    if (i < NN * 16) out[i] = selu_f(agg[i]);
}

__global__ __launch_bounds__(256)
void final_kernel(const float* __restrict__ agg, const float* __restrict__ x,
                  float* __restrict__ out) {
    int i = blockIdx.x * blockDim.x + threadIdx.x;
    if (i < NN * 16) out[i] = x[i] + 0.2f * selu_f(agg[i]);
}

// ---------------------------------------------------------------- edge conv
// One wave32 owns 32 edges: y = x[src] @ W2 via two 16-edge WMMA f32 tiles,
// then per-lane SELU-einsum + fp32 atomics into the (L2-resident) agg buffer.
template <int CIN>
__global__ __launch_bounds__(256)
void edge_kernel(const float* __restrict__ s0,
                 const float* __restrict__ s1,
                 const float* __restrict__ s2,
                 const int* __restrict__ srcI,
                 const int* __restrict__ dstI,
                 const float* __restrict__ feats,
                 const float* __restrict__ W0,   // (16,4) row-major
                 const float* __restrict__ b0,   // (16)
                 const float* __restrict__ W2,   // (CIN,16) row-major
                 const float* __restrict__ b2,   // (CIN)
                 float* __restrict__ agg) {
    constexpr int CH = CIN / 4;
    __shared__ float yws[8][32][17];   // stride 17: bank-conflict padding
    __shared__ float d2ws[8][2][32];

    const int lane = threadIdx.x & 31;
    const int w    = threadIdx.x >> 5;
    const int waveBase = (blockIdx.x * 8 + w) * 32;
    if (waveBase >= ETOT) return;      // wave-uniform: EXEC stays all-ones below
    const int Lm = lane & 15;
    const int hi = lane >> 4;          // 0: K pair {0,1}, 1: K pair {2,3}

    // A-matrix duty: lane loads rows M=Lm of tile0 (edges +0..15) and tile1 (+16..31)
    const int e0 = waveBase + Lm;
    const int e1 = e0 + 16;
    const int sn0 = (e0 >= ETOT) ? 0 : (e0 >= NEDGE ? (e0 - NEDGE) : srcI[e0]);
    const int sn1 = (e1 >= ETOT) ? 0 : (e1 >= NEDGE ? (e1 - NEDGE) : srcI[e1]);

    v8f acc0 = {0.f, 0.f, 0.f, 0.f, 0.f, 0.f, 0.f, 0.f};
    v8f acc1 = acc0;
    float p0 = 0.f, p1 = 0.f;          // partial xj . b2

#pragma unroll
    for (int ch = 0; ch < CH; ++ch) {
        const int kb  = 4 * ch + 2 * hi;                 // first K this lane handles
        const float* sArr = (kb < 16) ? s0 : ((kb < 32) ? s1 : s2);
        const int col = kb & 15;
        v2f a0, a1, bb;
        a0.x = sArr[sn0 * 16 + col];
        a0.y = sArr[sn0 * 16 + col + 1];
        a1.x = sArr[sn1 * 16 + col];
        a1.y = sArr[sn1 * 16 + col + 1];
        bb.x = W2[(kb + 0) * 16 + Lm];                   // B[k][n], n = Lm
        bb.y = W2[(kb + 1) * 16 + Lm];
        p0 = fmaf(a0.x, b2[kb], fmaf(a0.y, b2[kb + 1], p0));
        p1 = fmaf(a1.x, b2[kb], fmaf(a1.y, b2[kb + 1], p1));
        acc0 = __builtin_amdgcn_wmma_f32_16x16x4_f32(false, a0, false, bb,
                                                     (short)0, acc0, false, false);
        acc1 = __builtin_amdgcn_wmma_f32_16x16x4_f32(false, a1, false, bb,
                                                     (short)0, acc1, false, false);
    }

    // Stage D tiles: VGPR v, lanes 0-15 -> M=v, lanes 16-31 -> M=v+8; N = Lm.
#pragma unroll
    for (int v = 0; v < 8; ++v) {
        yws[w][v + 8 * hi][Lm]      = acc0[v];
        yws[w][16 + v + 8 * hi][Lm] = acc1[v];
    }
    d2ws[w][hi][Lm]      = p0;
    d2ws[w][hi][16 + Lm] = p1;

    // Cross-lane LDS dependency within the wave: drain DScnt explicitly.
    __builtin_amdgcn_wave_barrier();
    asm volatile("s_wait_dscnt 0" ::: "memory");
    __builtin_amdgcn_wave_barrier();

    // Einsum phase: one lane per edge.
    const int e = waveBase + lane;
    int dn = -1;
    float fx = 0.f, fy = 0.f, fz = 0.f;
    if (e < ETOT) {
        dn = (e >= NEDGE) ? (e - NEDGE) : dstI[e];
        fx = feats[e * 3 + 0];
        fy = feats[e * 3 + 1];
        fz = feats[e * 3 + 2];
    }
    float basev[16], wcv[16], yv[16];
#pragma unroll
    for (int h = 0; h < 16; ++h) {
        basev[h] = fmaf(fx, W0[h * 4 + 0],
                   fmaf(fy, W0[h * 4 + 1],
                   fmaf(fz, W0[h * 4 + 2], b0[h])));
        wcv[h] = W0[h * 4 + 3];
        yv[h]  = yws[w][lane][h];
    }
    const float dot2 = d2ws[w][0][lane] + d2ws[w][1][lane];

    if (dn >= 0) {
        float* aggRow = agg + dn * 16;
#pragma unroll 1
        for (int c = 0; c < 16; ++c) {
            float m = dot2;
            const float cf = (float)c;
#pragma unroll
            for (int h = 0; h < 16; ++h)
                m = fmaf(selu_f(fmaf(cf, wcv[h], basev[h])), yv[h], m);
            atomicAdd(aggRow + c, m);
        }
    }
}

// ---------------------------------------------------------------- launcher
extern "C" void kernel_launch(void* const* d_in, const int* in_sizes, int n_in,
                              void* d_out, int out_size, void* d_ws, size_t ws_size,
                              hipStream_t stream) {
    (void)in_sizes; (void)n_in; (void)out_size; (void)ws_size;

    const float* x     = (const float*)d_in[0];
    const int*   ei    = (const int*)d_in[1];
    const float* pos   = (const float*)d_in[2];
    const float* c1_W0 = (const float*)d_in[3];
    const float* c1_b0 = (const float*)d_in[4];
    const float* c1_W2 = (const float*)d_in[5];
    const float* c1_b2 = (const float*)d_in[6];
    const float* c1_bs = (const float*)d_in[7];
    const float* c2_W0 = (const float*)d_in[8];
    const float* c2_b0 = (const float*)d_in[9];
    const float* c2_W2 = (const float*)d_in[10];
    const float* c2_b2 = (const float*)d_in[11];
    const float* c2_bs = (const float*)d_in[12];
    const float* c3_W0 = (const float*)d_in[13];
    const float* c3_b0 = (const float*)d_in[14];
    const float* c3_W2 = (const float*)d_in[15];
    const float* c3_b2 = (const float*)d_in[16];
    const float* c3_bs = (const float*)d_in[17];

    const int* srcI = ei;
    const int* dstI = ei + NEDGE;

    float* feats = (float*)d_ws;                 // ETOT*3 floats
    float* x1    = feats + (size_t)ETOT * 3;     // NN*16
    float* x2    = x1 + NN * 16;                 // NN*16
    float* agg   = x2 + NN * 16;                 // NN*16
    float* out   = (float*)d_out;

    const int TB = 256;
    const int gE = (ETOT + TB - 1) / TB;         // also 256 edges per block
    const int gN = (NN * 16 + TB - 1) / TB;

    feats_kernel<<<gE, TB, 0, stream>>>(pos, srcI, dstI, feats);

    // layer 1: cin = 16
    init_agg_kernel<<<gN, TB, 0, stream>>>(agg, c1_bs);
    edge_kernel<16><<<gE, TB, 0, stream>>>(x, nullptr, nullptr, srcI, dstI, feats,
                                           c1_W0, c1_b0, c1_W2, c1_b2, agg);
    act_kernel<<<gN, TB, 0, stream>>>(agg, x1);

    // layer 2: cin = 32
    init_agg_kernel<<<gN, TB, 0, stream>>>(agg, c2_bs);
    edge_kernel<32><<<gE, TB, 0, stream>>>(x, x1, nullptr, srcI, dstI, feats,
                                           c2_W0, c2_b0, c2_W2, c2_b2, agg);
    act_kernel<<<gN, TB, 0, stream>>>(agg, x2);

    // layer 3: cin = 48, fused residual output
    init_agg_kernel<<<gN, TB, 0, stream>>>(agg, c3_bs);
    edge_kernel<48><<<gE, TB, 0, stream>>>(x, x1, x2, srcI, dstI, feats,
                                           c3_W0, c3_b0, c3_W2, c3_b2, agg);
    final_kernel<<<gN, TB, 0, stream>>>(agg, x, out);
}